// GCNNetUnmixing_75265006895410
// MI455X (gfx1250) — compile-verified
//
#include <hip/hip_runtime.h>

typedef float v2f __attribute__((ext_vector_type(2)));
typedef float v8f __attribute__((ext_vector_type(8)));

// ---------------------------------------------------------------------------
// Degree / normalization kernels
// ---------------------------------------------------------------------------
__global__ void deg_init_kernel(float* __restrict__ deg, int n) {
    int i = blockIdx.x * blockDim.x + threadIdx.x;
    if (i < n) deg[i] = 1.0f;   // self-loop weight 1.0 baked in
}

__global__ void deg_edges_kernel(const int* __restrict__ col,
                                 const float* __restrict__ ew,
                                 float* __restrict__ deg, int E) {
    int e = blockIdx.x * blockDim.x + threadIdx.x;
    if (e < E) atomicAdd(&deg[col[e]], ew[e]);
}

__global__ void deg_to_dinv_kernel(float* __restrict__ d, int n) {
    int i = blockIdx.x * blockDim.x + threadIdx.x;
    if (i < n) {
        float v = d[i];
        d[i] = (v > 0.0f) ? rsqrtf(v) : 0.0f;
    }
}

// ---------------------------------------------------------------------------
// FP32 WMMA GEMM: C[M,NN] = op(A[M,K]) @ B[K,NN]
//   op(A) = relu(A + bias_k) when FUSE_RELU_BIAS, else identity.
// Block = 256 threads = 8 waves sharing one 16-wide n-tile; B panel (K x 16)
// + bias staged in LDS once per block. Each wave computes MT=4 m-tiles
// (64 rows): every B fragment feeds 4 independent back-to-back WMMAs.
// Out-of-range rows are CLAMPED (they only affect C rows never stored), so
// the hot loop has zero exec divergence. Store epilogue has an unguarded
// fast path for full tiles (all but the last wave-row of the grid).
// ---------------------------------------------------------------------------
template <int K, int NN, int MT, bool FUSE_RELU_BIAS>
__global__ void gemm_wmma_f32_kernel(const float* __restrict__ A,
                                     const float* __restrict__ B,
                                     const float* __restrict__ bias,
                                     float* __restrict__ C, int M) {
    __shared__ float Bs[K * 16];
    __shared__ float bs[FUSE_RELU_BIAS ? K : 1];

    const int wave = threadIdx.x >> 5;
    const int lane = threadIdx.x & 31;
    const int half = lane >> 4;            // 0: K={0,1}, 1: K={2,3}
    const int sub  = lane & 15;
    const int n0   = blockIdx.y * 16;

    // Stage B panel [K x 16] (and bias) for this n-tile into LDS.
    for (int idx = threadIdx.x; idx < K * 16; idx += blockDim.x) {
        int kk = idx >> 4, nn = idx & 15;
        Bs[idx] = B[(long)kk * NN + (n0 + nn)];
    }
    if (FUSE_RELU_BIAS) {
        for (int idx = threadIdx.x; idx < K; idx += blockDim.x) bs[idx] = bias[idx];
    }
    __syncthreads();

    const int m_base = (blockIdx.x * (int)(blockDim.x >> 5) + wave) * (MT * 16);
    const int Mm1 = M - 1;

    long arow[MT];
#pragma unroll
    for (int t = 0; t < MT; ++t) {
        int r = m_base + t * 16 + sub;
        r = (r < Mm1) ? r : Mm1;           // clamp instead of mask
        arow[t] = (long)r * K;
    }

    v8f acc[MT];
#pragma unroll
    for (int t = 0; t < MT; ++t) acc[t] = (v8f){0.f,0.f,0.f,0.f,0.f,0.f,0.f,0.f};

    for (int k = 0; k < K; k += 4) {
        const int ka = k + 2 * half;
        v2f b;
        b.x = Bs[ka * 16 + sub];           // ds_load_2addr, conflict-free
        b.y = Bs[(ka + 1) * 16 + sub];
        float bk0 = 0.f, bk1 = 0.f;
        if (FUSE_RELU_BIAS) { bk0 = bs[ka]; bk1 = bs[ka + 1]; }
#pragma unroll
        for (int t = 0; t < MT; ++t) {
            v2f a;
            float a0 = A[arow[t] + ka];
            float a1 = A[arow[t] + ka + 1];
            if (FUSE_RELU_BIAS) {
                a0 = fmaxf(a0 + bk0, 0.0f);
                a1 = fmaxf(a1 + bk1, 0.0f);
            }
            a.x = a0; a.y = a1;
            acc[t] = __builtin_amdgcn_wmma_f32_16x16x4_f32(
                false, a, false, b, (short)0, acc[t], false, false);
        }
    }

    // Epilogue. C layout: VGPR r holds row (r + 8*half) of the tile.
    if (m_base + MT * 16 <= M) {
        // Fast path: full tiles, no per-row guards.
#pragma unroll
        for (int t = 0; t < MT; ++t) {
            float* cp = C + (long)(m_base + t * 16 + 8 * half) * NN + (n0 + sub);
#pragma unroll
            for (int r = 0; r < 8; ++r) cp[(long)r * NN] = acc[t][r];
        }
    } else {
        // Boundary wave-row only.
#pragma unroll
        for (int t = 0; t < MT; ++t) {
#pragma unroll
            for (int r = 0; r < 8; ++r) {
                int mr = m_base + t * 16 + r + 8 * half;
                if (mr < M) C[(long)mr * NN + (n0 + sub)] = acc[t][r];
            }
        }
    }
}

// ---------------------------------------------------------------------------
// Aggregation: out[i,:] starts as self-loop term dinv[i]^2 * h[i,:] [+ bias],
// then edges atomically accumulate norm * h[row,:] into out[col,:].
// Streaming paths use float4 (global_load/store_b128); gathers are one b128
// per lane; scatter is global_atomic_add_f32 (both operands L2-resident:
// h and out are each <= 51.2 MB vs 192 MB L2).
// ---------------------------------------------------------------------------
template <int F, bool ADD_BIAS>
__global__ void agg_init_kernel(const float* __restrict__ h,
                                const float* __restrict__ dinv,
                                const float* __restrict__ bias,
                                float* __restrict__ out, long total4) {
    long i = (long)blockIdx.x * blockDim.x + threadIdx.x;   // float4 index
    if (i >= total4) return;
    long base = i * 4;
    long node = base / F;
    int  f    = (int)(base % F);
    float d = dinv[node];
    float s = d * d;
    float4 v = ((const float4*)h)[i];
    v.x *= s; v.y *= s; v.z *= s; v.w *= s;
    if (ADD_BIAS) {
        v.x += bias[f];     v.y += bias[f + 1];
        v.z += bias[f + 2]; v.w += bias[f + 3];
    }
    ((float4*)out)[i] = v;
}

template <int F>
__global__ void agg_edges_kernel(const int* __restrict__ row,
                                 const int* __restrict__ col,
                                 const float* __restrict__ ew,
                                 const float* __restrict__ dinv,
                                 const float* __restrict__ h,
                                 float* __restrict__ out, int E) {
    constexpr int G = F / 4;               // lanes per edge, float4 each
    long t = (long)blockIdx.x * blockDim.x + threadIdx.x;
    long e = t / G;
    int  g = (int)(t % G);
    if (e >= E) return;
    int r = row[e], c = col[e];
    float norm = dinv[r] * ew[e] * dinv[c];
    float4 hv = *(const float4*)(h + (long)r * F + 4 * g);   // one b128 gather
    float* oc = out + (long)c * F + 4 * g;
    atomicAdd(&oc[0], norm * hv.x);
    atomicAdd(&oc[1], norm * hv.y);
    atomicAdd(&oc[2], norm * hv.z);
    atomicAdd(&oc[3], norm * hv.w);
}

// ---------------------------------------------------------------------------
__global__ void relu4_kernel(const float4* __restrict__ in,
                             float4* __restrict__ out, long n4) {
    long i = (long)blockIdx.x * blockDim.x + threadIdx.x;
    if (i >= n4) return;
    float4 v = in[i];
    v.x = fmaxf(v.x, 0.f); v.y = fmaxf(v.y, 0.f);
    v.z = fmaxf(v.z, 0.f); v.w = fmaxf(v.w, 0.f);
    out[i] = v;
}

// ---------------------------------------------------------------------------
extern "C" void kernel_launch(void* const* d_in, const int* in_sizes, int n_in,
                              void* d_out, int out_size, void* d_ws, size_t ws_size,
                              hipStream_t stream) {
    constexpr int IN_DIM = 200, HID = 128, OUT = 16, MT = 4;

    const float* x   = (const float*)d_in[0];
    const int*   ei  = (const int*)  d_in[1];
    const float* ew  = (const float*)d_in[2];
    const float* W1  = (const float*)d_in[3];
    const float* b1  = (const float*)d_in[4];
    const float* W2  = (const float*)d_in[5];
    const float* b2  = (const float*)d_in[6];
    const float* abA = (const float*)d_in[7];
    const float* emS = (const float*)d_in[8];

    const int E  = in_sizes[2];                 // 3,200,000
    const int N  = in_sizes[7] / OUT;           // 100,000 nodes
    const int ES = in_sizes[8];                 // 200
    const int* erow = ei;
    const int* ecol = ei + E;

    // Workspace layout (floats): ~110 MB
    float* ws   = (float*)d_ws;
    float* dinv = ws;                           // N
    float* h1   = dinv + N;                     // N * HID
    float* agg1 = h1 + (size_t)N * HID;         // N * HID
    float* h2   = agg1 + (size_t)N * HID;       // N * OUT

    float* out_h  = (float*)d_out;              // N * OUT
    float* out_ab = out_h + (size_t)N * OUT;    // N * OUT
    float* out_es = out_ab + (size_t)N * OUT;   // ES

    const int T = 256;
    auto blocks = [](long n, int t) { return (unsigned)((n + t - 1) / t); };

    // 1) degree -> dinv (self-loop weight 1.0 baked into init)
    deg_init_kernel<<<blocks(N, T), T, 0, stream>>>(dinv, N);
    deg_edges_kernel<<<blocks(E, T), T, 0, stream>>>(ecol, ew, dinv, E);
    deg_to_dinv_kernel<<<blocks(N, T), T, 0, stream>>>(dinv, N);

    // Rows covered per block: 8 waves * MT tiles * 16 rows = 512
    const unsigned gx = blocks(N, 8 * MT * 16);

    // 2) h1 = x @ W1  (fp32 WMMA, B panel staged in LDS)
    gemm_wmma_f32_kernel<IN_DIM, HID, MT, false>
        <<<dim3(gx, HID / 16), T, 0, stream>>>(x, W1, nullptr, h1, N);

    // 3) agg1 = dinv^2*h1 (self loops) + scatter(norm * h1[row] -> col)
    agg_init_kernel<HID, false><<<blocks((long)N * HID / 4, T), T, 0, stream>>>(
        h1, dinv, nullptr, agg1, (long)N * HID / 4);
    agg_edges_kernel<HID><<<blocks((long)E * (HID / 4), T), T, 0, stream>>>(
        erow, ecol, ew, dinv, h1, agg1, E);

    // 4) h2 = relu(agg1 + b1) @ W2  (bias+relu fused into A-fragment load)
    gemm_wmma_f32_kernel<HID, OUT, MT, true>
        <<<dim3(gx, OUT / 16), T, 0, stream>>>(agg1, W2, b1, h2, N);

    // 5) out_h = dinv^2*h2 + b2 (self loops + bias) + edge scatter
    agg_init_kernel<OUT, true><<<blocks((long)N * OUT / 4, T), T, 0, stream>>>(
        h2, dinv, b2, out_h, (long)N * OUT / 4);
    agg_edges_kernel<OUT><<<blocks((long)E * (OUT / 4), T), T, 0, stream>>>(
        erow, ecol, ew, dinv, h2, out_h, E);

    // 6) remaining outputs: relu(abundanceA), relu(endmemberS)
    relu4_kernel<<<blocks((long)N * OUT / 4, T), T, 0, stream>>>(
        (const float4*)abA, (float4*)out_ab, (long)N * OUT / 4);
    relu4_kernel<<<blocks(ES / 4, T), T, 0, stream>>>(
        (const float4*)emS, (float4*)out_es, ES / 4);
}